// FullCliffordMeanField3DClassifier_13950053777565
// MI455X (gfx1250) — compile-verified
//
#include <hip/hip_runtime.h>
#include <math.h>

typedef __attribute__((ext_vector_type(2))) float v2f;
typedef __attribute__((ext_vector_type(8))) float v8f;

#define NROWS 4096
#define NBATCH 1024
#define HDIM 64

__device__ __forceinline__ float gelu_exact(float x) {
    return 0.5f * x * (1.0f + erff(x * 0.70710678118654752440f));
}

// ---------------------------------------------------------------------------
// Kernel 1: per-batch streaming reduction over N=4096 rows of 6 floats.
// Computes sums of: dot(p,v), (p0v1-p1v0), (p0v2-p2v0), (p1v2-p2v1), v0,v1,v2
// One block (256 threads) per batch; rows read in groups of 4 (= six float4).
// ---------------------------------------------------------------------------
__global__ __launch_bounds__(256) void clifford_reduce_kernel(
    const float* __restrict__ x, float* __restrict__ ws) {
    const int b   = blockIdx.x;
    const int tid = threadIdx.x;
    const float4* __restrict__ p4 =
        (const float4*)(x + (size_t)b * (size_t)NROWS * 6);

    float acc[7];
#pragma unroll
    for (int i = 0; i < 7; ++i) acc[i] = 0.0f;

    auto accrow = [&](float p0, float p1, float p2,
                      float v0, float v1, float v2) {
        acc[0] += p0 * v0 + p1 * v1 + p2 * v2;   // scalar (dot)
        acc[1] += p0 * v1 - p1 * v0;             // e12
        acc[2] += p0 * v2 - p2 * v0;             // e13
        acc[3] += p1 * v2 - p2 * v1;             // e23
        acc[4] += v0;
        acc[5] += v1;
        acc[6] += v2;
    };

    // 4096 rows -> 1024 groups of 4 rows; 256 threads x 4 groups each
#pragma unroll
    for (int i = 0; i < 4; ++i) {
        const int g = tid + 256 * i;
        const float4 f0 = p4[g * 6 + 0];
        const float4 f1 = p4[g * 6 + 1];
        const float4 f2 = p4[g * 6 + 2];
        const float4 f3 = p4[g * 6 + 3];
        const float4 f4 = p4[g * 6 + 4];
        const float4 f5 = p4[g * 6 + 5];
        accrow(f0.x, f0.y, f0.z, f0.w, f1.x, f1.y);   // row 0
        accrow(f1.z, f1.w, f2.x, f2.y, f2.z, f2.w);   // row 1
        accrow(f3.x, f3.y, f3.z, f3.w, f4.x, f4.y);   // row 2
        accrow(f4.z, f4.w, f5.x, f5.y, f5.z, f5.w);   // row 3
    }

    __shared__ float sm[256][8];
#pragma unroll
    for (int c = 0; c < 7; ++c) sm[tid][c] = acc[c];
    __syncthreads();

#pragma unroll
    for (int s = 128; s > 0; s >>= 1) {
        if (tid < s) {
#pragma unroll
            for (int c = 0; c < 7; ++c) sm[tid][c] += sm[tid + s][c];
        }
        __syncthreads();
    }
    if (tid == 0) {
#pragma unroll
        for (int c = 0; c < 7; ++c) ws[b * 8 + c] = sm[0][c];
    }
}

// ---------------------------------------------------------------------------
// Kernel 2: per-batch geometric product GP(mean_vel, mean_pv) via the exact
// Cayley sign rule of the reference, then the 8->64->64->2 MLP using
// V_WMMA_F32_16X16X4_F32 tiles (fp32 matrix pipe, full precision).
// One wave (32 threads) handles 16 batch rows. Grid = 1024/16 = 64 blocks.
// ---------------------------------------------------------------------------
__global__ __launch_bounds__(32) void clifford_mlp_kernel(
    const float* __restrict__ ws,
    const float* __restrict__ W1, const float* __restrict__ b1,
    const float* __restrict__ W2, const float* __restrict__ b2,
    const float* __restrict__ W3, const float* __restrict__ b3,
    float* __restrict__ out) {
    const int lane = threadIdx.x;
    const int row0 = blockIdx.x * 16;

    __shared__ float mv[16][8];     // mean_mv tile (A input of stage 1)
    __shared__ float h[16][64];     // hidden activations (reused h1 -> h2)

    // ---- per-batch geometric product (lanes 0..15, one batch each) ----
    if (lane < 16) {
        const int bi = row0 + lane;
        const float invN = 1.0f / (float)NROWS;
        float u[8], v[8], r[8];
#pragma unroll
        for (int i = 0; i < 8; ++i) { u[i] = 0.0f; v[i] = 0.0f; r[i] = 0.0f; }
        // u = embedded mean velocity (grade-1 slots 1..3)
        u[1] = ws[bi * 8 + 4] * invN;
        u[2] = ws[bi * 8 + 5] * invN;
        u[3] = ws[bi * 8 + 6] * invN;
        // v = mean_pv: scalar + bivectors e12,e13,e23 (slots 0,4,5,6)
        v[0] = ws[bi * 8 + 0] * invN;
        v[4] = ws[bi * 8 + 1] * invN;
        v[5] = ws[bi * 8 + 2] * invN;
        v[6] = ws[bi * 8 + 3] * invN;

        // Cayley table of Cl(3,0), identical algorithm to the reference.
        const int bits[8] = {0, 1, 2, 4, 3, 5, 6, 7};
        int inv[8];
#pragma unroll
        for (int i = 0; i < 8; ++i) inv[bits[i]] = i;
        for (int a = 0; a < 8; ++a) {
            for (int c = 0; c < 8; ++c) {
                const int ba = bits[a], bc = bits[c];
                int s = 1, aa = ba >> 1;
                while (aa) {
                    if (__popc(aa & bc) & 1) s = -s;
                    aa >>= 1;
                }
                r[inv[ba ^ bc]] += (float)s * u[a] * v[c];
            }
        }
#pragma unroll
        for (int k = 0; k < 8; ++k) mv[lane][k] = r[k];
    }
    __syncthreads();

    // WMMA fragment coordinates (wave32, 16x16x4 f32):
    //   A (16x4):  lane&15 = row M, (lane>>4)*2 = first K in v2f
    //   B (4x16):  lane&15 = col N, (lane>>4)*2 = first K in v2f
    //   C/D:       VGPR r holds M = r + 8*(lane>>4), N = lane&15
    const int half = lane >> 4;
    const int mrow = lane & 15;
    const int ncl  = lane & 15;
    const int kb   = half * 2;

    // ---------------- stage 1: [16x8] @ W1[8x64] + b1, GELU ----------------
#pragma unroll
    for (int j = 0; j < 4; ++j) {
        const int ncol = j * 16 + ncl;
        const float bias = b1[ncol];
        v8f c;
#pragma unroll
        for (int rI = 0; rI < 8; ++rI) c[rI] = bias;
#pragma unroll
        for (int s = 0; s < 2; ++s) {
            const int k0 = s * 4 + kb;
            v2f a, bb;
            a.x  = mv[mrow][k0];
            a.y  = mv[mrow][k0 + 1];
            bb.x = W1[(k0)     * HDIM + ncol];
            bb.y = W1[(k0 + 1) * HDIM + ncol];
            c = __builtin_amdgcn_wmma_f32_16x16x4_f32(
                    false, a, false, bb, (short)0, c, false, false);
        }
#pragma unroll
        for (int rI = 0; rI < 8; ++rI)
            h[rI + 8 * half][ncol] = gelu_exact(c[rI]);
    }
    __syncthreads();

    // ---------------- stage 2: [16x64] @ W2[64x64] + b2, GELU --------------
    v8f acc2[4];
#pragma unroll
    for (int j = 0; j < 4; ++j) {
        const float bias = b2[j * 16 + ncl];
#pragma unroll
        for (int rI = 0; rI < 8; ++rI) acc2[j][rI] = bias;
    }
#pragma unroll
    for (int s = 0; s < 16; ++s) {
        const int k0 = s * 4 + kb;
        v2f a;
        a.x = h[mrow][k0];
        a.y = h[mrow][k0 + 1];
#pragma unroll
        for (int j = 0; j < 4; ++j) {
            const int ncol = j * 16 + ncl;
            v2f bb;
            bb.x = W2[(k0)     * HDIM + ncol];
            bb.y = W2[(k0 + 1) * HDIM + ncol];
            acc2[j] = __builtin_amdgcn_wmma_f32_16x16x4_f32(
                          false, a, false, bb, (short)0, acc2[j], false, false);
        }
    }
    __syncthreads();
#pragma unroll
    for (int j = 0; j < 4; ++j) {
        const int ncol = j * 16 + ncl;
#pragma unroll
        for (int rI = 0; rI < 8; ++rI)
            h[rI + 8 * half][ncol] = gelu_exact(acc2[j][rI]);
    }
    __syncthreads();

    // ------------- stage 3: [16x64] @ W3[64x2] + b3 (N padded to 16) -------
    v8f c3;
    const float bias3 = (ncl < 2) ? b3[ncl] : 0.0f;
#pragma unroll
    for (int rI = 0; rI < 8; ++rI) c3[rI] = bias3;
#pragma unroll
    for (int s = 0; s < 16; ++s) {
        const int k0 = s * 4 + kb;
        v2f a, bb;
        a.x  = h[mrow][k0];
        a.y  = h[mrow][k0 + 1];
        bb.x = (ncl < 2) ? W3[(k0)     * 2 + ncl] : 0.0f;
        bb.y = (ncl < 2) ? W3[(k0 + 1) * 2 + ncl] : 0.0f;
        c3 = __builtin_amdgcn_wmma_f32_16x16x4_f32(
                 false, a, false, bb, (short)0, c3, false, false);
    }
    if (ncl < 2) {
#pragma unroll
        for (int rI = 0; rI < 8; ++rI)
            out[(row0 + rI + 8 * half) * 2 + ncl] = c3[rI];
    }
}

// ---------------------------------------------------------------------------
extern "C" void kernel_launch(void* const* d_in, const int* in_sizes, int n_in,
                              void* d_out, int out_size, void* d_ws, size_t ws_size,
                              hipStream_t stream) {
    const float* x  = (const float*)d_in[0];
    const float* W1 = (const float*)d_in[1];
    const float* b1 = (const float*)d_in[2];
    const float* W2 = (const float*)d_in[3];
    const float* b2 = (const float*)d_in[4];
    const float* W3 = (const float*)d_in[5];
    const float* b3 = (const float*)d_in[6];
    float* out = (float*)d_out;
    float* ws  = (float*)d_ws;   // 1024 batches x 8 floats = 32 KB, first 7 of each 8 written

    clifford_reduce_kernel<<<NBATCH, 256, 0, stream>>>(x, ws);
    clifford_mlp_kernel<<<NBATCH / 16, 32, 0, stream>>>(ws, W1, b1, W2, b2, W3, b3, out);
}